// MatchNet_11536282157759
// MI455X (gfx1250) — compile-verified
//
#include <hip/hip_runtime.h>
#include <math.h>

typedef __attribute__((ext_vector_type(2))) float v2f;
typedef __attribute__((ext_vector_type(8))) float v8f;

#define MASKV (-4294967295.0f)

__device__ __forceinline__ float sigf(float x){ return 1.0f/(1.0f+expf(-x)); }

// ---------------------------------------------------------------------------
// Kernel 0: zero-padded Whh copy  Wpad[dir][n(608)][k(152)]  (pad n>=600,k>=150)
// Removes every guard from the serial recurrence's WMMA inner loop.
// ---------------------------------------------------------------------------
__global__ __launch_bounds__(256) void k0_padwhh(const float* __restrict__ WhhF,
        const float* __restrict__ WhhB, float* __restrict__ Wpad)
{
    int idx = blockIdx.x*256 + threadIdx.x;
    if (idx >= 2*608*152) return;
    int dir = idx / (608*152);
    int rem = idx - dir*(608*152);
    int n = rem / 152, k = rem - n*152;
    const float* Whh = dir ? WhhB : WhhF;
    Wpad[idx] = (n < 600 && k < 150) ? Whh[n*150+k] : 0.0f;
}

// ---------------------------------------------------------------------------
// Kernel 1: per (b,s) attention + masked softmax over i (q_att = "check"),
// C_d[j,k] = sum_i q_att[i,j]*C_s[i,k], write time-masked LSTM input X.
// ---------------------------------------------------------------------------
__global__ __launch_bounds__(256) void k1_att(const float* __restrict__ Cs,
        const int* __restrict__ Clen, const int* __restrict__ Cslen,
        const float* __restrict__ Rs, const int* __restrict__ Rlen,
        float* __restrict__ check, float* __restrict__ X)
{
    const int s = blockIdx.x, b = blockIdx.y;
    const int tid = threadIdx.x;
    __shared__ float rbuf[4*300];
    __shared__ float lg[128*4];
    __shared__ float qmax[4], qsum[4];
    const float* cs = Cs + ((size_t)(b*64+s))*128*300;
    for (int e=tid; e<1200; e+=256) rbuf[e] = Rs[(size_t)b*1200 + e];
    __syncthreads();
    const int ilen = Cslen[b*64+s];
    const int jlen = Rlen[b];
    if (tid < 128) {
        float a0=0.f,a1=0.f,a2=0.f,a3=0.f;
        const float* row = cs + tid*300;
        for (int k=0;k<300;k++){
            float x = row[k];
            a0 += x*rbuf[k]; a1 += x*rbuf[300+k];
            a2 += x*rbuf[600+k]; a3 += x*rbuf[900+k];
        }
        float acc[4] = {a0,a1,a2,a3};
        for (int j=0;j<4;j++){
            bool mk = (tid < ilen) && (j < jlen);
            lg[tid*4+j] = mk ? acc[j] : MASKV;   // att*mask + MASK*(1-mask)
        }
    }
    __syncthreads();
    if (tid < 4) {                                // softmax over i (axis=-2)
        float mx = -3.4e38f;
        for (int i=0;i<128;i++) mx = fmaxf(mx, lg[i*4+tid]);
        float sm = 0.f;
        for (int i=0;i<128;i++) sm += expf(lg[i*4+tid]-mx);
        qmax[tid]=mx; qsum[tid]=sm;
    }
    __syncthreads();
    if (tid < 128) {
        for (int j=0;j<4;j++){
            float q = expf(lg[tid*4+j]-qmax[j]) / qsum[j];
            lg[tid*4+j] = q;
            check[(((size_t)(b*64+s))*128 + tid)*4 + j] = q;
        }
    }
    __syncthreads();
    const float tm = (s < Clen[b]) ? 1.0f : 0.0f; // mask x before the LSTM
    for (int k=tid; k<300; k+=256) {
        float a0=0.f,a1=0.f,a2=0.f,a3=0.f;
        for (int i=0;i<128;i++){
            float x = cs[i*300+k];
            a0 += lg[i*4+0]*x; a1 += lg[i*4+1]*x;
            a2 += lg[i*4+2]*x; a3 += lg[i*4+3]*x;
        }
        float a[4] = {a0,a1,a2,a3};
        for (int j=0;j<4;j++)
            X[(((size_t)(b*4+j))*64 + s)*300 + k] = a[j]*tm;
    }
}

// ---------------------------------------------------------------------------
// Kernel 2: G = X @ Wih^T + (bih+bhh) for both directions with fp32 WMMA.
// M=8192 rows, N padded to 608 per dir (76 N-tiles total), K=300 (75 steps).
// Block: 8 waves x one 16-wide N tile; B tile staged in LDS.
// ---------------------------------------------------------------------------
__global__ __launch_bounds__(256) void k2_ingemm(const float* __restrict__ X,
        const float* __restrict__ WihF, const float* __restrict__ WihB,
        const float* __restrict__ bihF, const float* __restrict__ bhhF,
        const float* __restrict__ bihB, const float* __restrict__ bhhB,
        float* __restrict__ Gf, float* __restrict__ Gb)
{
    const int ntile = blockIdx.y;          // 0..75 ; dir boundary at tile 38
    const int dir = ntile / 38;
    const int tid = threadIdx.x;
    const int w = tid >> 5, lane = tid & 31;
    __shared__ float bt[300*16];           // B(k,n) tile, k-major
    const float* Wih = dir ? WihB : WihF;
    for (int e=tid; e<4800; e+=256){
        int k = e>>4, nl = e&15;
        int nn = ntile*16 + nl - dir*608;
        bt[e] = (nn < 600) ? Wih[nn*300 + k] : 0.0f;
    }
    __syncthreads();
    const int mloc = lane & 15;            // A: M row  /  B: N col
    const int mg = (blockIdx.x*8 + w)*16 + mloc;
    const int kofs = (lane < 16) ? 0 : 2;  // lanes 16-31 hold K+2,K+3
    const float* xr = X + (size_t)mg*300;
    v8f acc = {};
    for (int kt=0; kt<75; kt++){
        int klo = kt*4 + kofs;
        v2f a, bb;
        a.x  = xr[klo];              a.y  = xr[klo+1];
        bb.x = bt[klo*16 + mloc];    bb.y = bt[(klo+1)*16 + mloc];
        acc = __builtin_amdgcn_wmma_f32_16x16x4_f32(false, a, false, bb,
                                                    (short)0, acc, false, false);
    }
    const int nl = lane & 15;
    const int nn = ntile*16 + nl - dir*608;
    if (nn < 600) {
        float bias = dir ? (bihB[nn]+bhhB[nn]) : (bihF[nn]+bhhF[nn]);
        float* G = dir ? Gb : Gf;
        int mbase = (blockIdx.x*8 + w)*16 + ((lane >= 16) ? 8 : 0);
        for (int v=0; v<8; v++)
            G[(size_t)(mbase+v)*600 + nn] = acc[v] + bias;
    }
}

// ---------------------------------------------------------------------------
// Kernel 3: LSTM recurrence. 16 independent blocks = 8 row-groups x 2 dirs.
// Gate GEMM h @ Whh^T via guard-free WMMA on the zero-padded Whh (K=152,
// N=608); G added during the elementwise update (coalesced); next step's G
// prefetched during the update phase. Output masked by C_len.
// ---------------------------------------------------------------------------
__global__ __launch_bounds__(256) void k3_lstm(const float* __restrict__ Gf,
        const float* __restrict__ Gb, const float* __restrict__ Wpad,
        const int* __restrict__ Clen, float* __restrict__ H)
{
    const int dir = blockIdx.x >> 3;
    const int r0  = (blockIdx.x & 7) * 16;
    const int tid = threadIdx.x;
    const int w = tid >> 5, lane = tid & 31;
    const float* G  = dir ? Gb : Gf;
    const float* wp = Wpad + (size_t)dir*(608*152);
    __shared__ float hbuf[16*152];   // h state, K padded to 152 (cols 150/151 = 0)
    __shared__ float cst [16*152];   // c state
    __shared__ float gbuf[16*608];   // gate GEMM accumulator
    for (int e=tid; e<16*152; e+=256){ hbuf[e]=0.f; cst[e]=0.f; }
    __syncthreads();
    const int mloc = lane & 15;
    const int kofs = (lane < 16) ? 0 : 2;
    for (int tau=0; tau<64; tau++){
        const int t_in = dir ? (63-tau) : tau;
        for (int ntile=w; ntile<38; ntile+=8){
            const int n = ntile*16 + mloc;
            const v2f* wcol = (const v2f*)(wp + (size_t)n*152 + kofs);
            const v2f* hrow = (const v2f*)(hbuf + mloc*152 + kofs);
            v8f acc = {};
            #pragma unroll
            for (int kt=0; kt<38; kt++){
                v2f a  = hrow[kt*2];      // h[mloc][klo], h[mloc][klo+1]
                v2f bb = wcol[kt*2];      // Wpad[n][klo], Wpad[n][klo+1]
                acc = __builtin_amdgcn_wmma_f32_16x16x4_f32(false, a, false, bb,
                                                            (short)0, acc, false, false);
            }
            if (n < 600){
                int mb = (lane >= 16) ? 8 : 0;
                #pragma unroll
                for (int v=0; v<8; v++)
                    gbuf[(mb+v)*608 + n] = acc[v];
            }
        }
        __syncthreads();
        {   // prefetch next timestep's gate inputs (global_prefetch_b8)
            int tn = dir ? (t_in > 0 ? t_in-1 : 0) : (t_in < 63 ? t_in+1 : 63);
            int m = tid >> 4, seg = tid & 15;
            __builtin_prefetch(&G[(((size_t)(r0+m))*64 + tn)*600 + seg*38], 0, 3);
        }
        for (int e=tid; e<16*150; e+=256){
            int m = e/150, hh = e - m*150;
            size_t gb = (((size_t)(r0+m))*64 + t_in)*600;
            float i_ = gbuf[m*608 +       hh] + G[gb +       hh];
            float f_ = gbuf[m*608 + 150 + hh] + G[gb + 150 + hh];
            float g_ = gbuf[m*608 + 300 + hh] + G[gb + 300 + hh];
            float o_ = gbuf[m*608 + 450 + hh] + G[gb + 450 + hh];
            float c = sigf(f_)*cst[m*152+hh] + sigf(i_)*tanhf(g_);
            cst[m*152+hh] = c;
            float h = sigf(o_)*tanhf(c);
            hbuf[m*152+hh] = h;
            int bj = r0 + m;
            float outv = (t_in < Clen[bj>>2]) ? h : 0.0f;   // H * mask
            H[(((size_t)bj)*64 + t_in)*300 + dir*150 + hh] = outv;
        }
        __syncthreads();
    }
}

// ---------------------------------------------------------------------------
// Kernel 4a: att2[b,j,s] = <H[b,j,s,:], R_h[b,j,:]>
// ---------------------------------------------------------------------------
__global__ __launch_bounds__(256) void k4a_att2(const float* __restrict__ H,
        const float* __restrict__ Rh, float* __restrict__ att2)
{
    const int bj = blockIdx.x;
    const int tid = threadIdx.x;
    const int s = tid >> 2, part = tid & 3;
    __shared__ float red[256];
    const float* h = H  + (((size_t)bj)*64 + s)*300 + part*75;
    const float* r = Rh + (size_t)bj*300 + part*75;
    float acc = 0.f;
    for (int k=0;k<75;k++) acc += h[k]*r[k];
    red[tid] = acc;
    __syncthreads();
    if (part == 0)
        att2[bj*64 + s] = red[tid] + red[tid+1] + red[tid+2] + red[tid+3];
}

// ---------------------------------------------------------------------------
// Kernel 4b: masked softmax over s, C_h, (C_h-R_h)@simW^T norms -> dist,
// and the iterative argmax/diff loop -> all_diff. One block per b.
// ---------------------------------------------------------------------------
__global__ __launch_bounds__(256) void k4b_final(const float* __restrict__ att2,
        const float* __restrict__ H, const float* __restrict__ Rh,
        const float* __restrict__ simW, const int* __restrict__ Clen,
        const int* __restrict__ Rlen,
        float* __restrict__ dist, float* __restrict__ alldiff)
{
    const int b = blockIdx.x;
    const int tid = threadIdx.x;
    __shared__ float a2[256];       // raw att2[b, j, s]
    __shared__ float p2[256];       // softmax over s
    __shared__ float u[1200];       // C_h - R_h  [4,300]
    __shared__ float part[256*4];
    a2[tid] = att2[b*256 + tid];
    __syncthreads();
    const int cl = Clen[b], jl = Rlen[b];
    if (tid < 4){
        int jj = tid;
        float mx = -3.4e38f;
        for (int s=0;s<64;s++){
            float l = (jj<jl && s<cl) ? a2[jj*64+s] : MASKV;
            mx = fmaxf(mx, l);
        }
        float sm = 0.f;
        for (int s=0;s<64;s++){
            float l = (jj<jl && s<cl) ? a2[jj*64+s] : MASKV;
            float e = expf(l-mx);
            p2[jj*64+s] = e; sm += e;
        }
        for (int s=0;s<64;s++) p2[jj*64+s] /= sm;
    }
    __syncthreads();
    for (int e=tid; e<1200; e+=256){
        int j = e/300, k = e - j*300;
        float acc = 0.f;
        for (int s=0;s<64;s++)
            acc += p2[j*64+s] * H[(((size_t)(b*4+j))*64+s)*300 + k];
        u[e] = acc - Rh[(size_t)(b*4+j)*300 + k];
    }
    __syncthreads();
    float ln[4] = {0.f,0.f,0.f,0.f};
    for (int e=tid; e<1200; e+=256){
        int j = e/300, n = e - j*300;
        float y = 0.f;
        for (int k=0;k<300;k++) y += u[j*300+k]*simW[n*300+k];
        ln[j] += y*y;
    }
    for (int j=0;j<4;j++) part[tid*4+j] = ln[j];
    __syncthreads();
    if (tid == 0){
        float nr[4]={0.f,0.f,0.f,0.f};
        for (int t=0;t<256;t++)
            for (int j=0;j<4;j++) nr[j] += part[t*4+j];
        dist[b] = -(sqrtf(nr[0])+sqrtf(nr[1])+sqrtf(nr[2])+sqrtf(nr[3]));
        // iterative argmax / masked-diff loop (first-occurrence argmax)
        float smax[4]; int sidx[4];
        for (int j=0;j<4;j++){
            float mx = a2[j*64+0]; int mi = 0;
            for (int s=1;s<64;s++){ float v=a2[j*64+s]; if (v>mx){mx=v;mi=s;} }
            smax[j]=mx; sidx[j]=mi;
        }
        float best = 3.4e38f;
        for (int t=0;t<3;t++){
            float pm = 3.4e38f;
            for (int s=0;s<64;s++){
                float m = (s < sidx[t]) ? 1.0f : 0.0f;
                for (int jp=t+1; jp<4; jp++){
                    float d = (smax[t] - a2[jp*64+s]) * m;
                    pm = fminf(pm, d);
                }
            }
            best = fminf(best, pm);
        }
        alldiff[b] = best;
    }
}

// ---------------------------------------------------------------------------
extern "C" void kernel_launch(void* const* d_in, const int* in_sizes, int n_in,
                              void* d_out, int out_size, void* d_ws, size_t ws_size,
                              hipStream_t stream)
{
    const float* Cs    = (const float*)d_in[0];
    const int*   Clen  = (const int*)  d_in[1];
    const int*   Cslen = (const int*)  d_in[2];
    const float* Rs    = (const float*)d_in[3];
    const float* Rh    = (const float*)d_in[4];
    const int*   Rlen  = (const int*)  d_in[5];
    const float* simW  = (const float*)d_in[6];
    const float* WihF  = (const float*)d_in[7];
    const float* WhhF  = (const float*)d_in[8];
    const float* bihF  = (const float*)d_in[9];
    const float* bhhF  = (const float*)d_in[10];
    const float* WihB  = (const float*)d_in[11];
    const float* WhhB  = (const float*)d_in[12];
    const float* bihB  = (const float*)d_in[13];
    const float* bhhB  = (const float*)d_in[14];

    float* out     = (float*)d_out;
    float* dist    = out;          // [32]
    float* alldiff = out + 32;     // [32]
    float* check   = out + 64;     // [32,64,128,4]

    float* ws   = (float*)d_ws;
    float* X    = ws;               // [128,64,300]  masked LSTM input
    float* Gf   = X  + 2457600;     // [8192,600]
    float* Gb   = Gf + 4915200;     // [8192,600]
    float* H    = Gb + 4915200;     // [128,64,300]  BiLSTM output (masked)
    float* att2 = H  + 2457600;     // [32,4,64]
    float* Wpad = att2 + 8192;      // [2,608,152]  zero-padded Whh

    k0_padwhh<<<722,        256, 0, stream>>>(WhhF, WhhB, Wpad);
    k1_att   <<<dim3(64,32), 256, 0, stream>>>(Cs, Clen, Cslen, Rs, Rlen, check, X);
    k2_ingemm<<<dim3(64,76), 256, 0, stream>>>(X, WihF, WihB, bihF, bhhF, bihB, bhhB, Gf, Gb);
    k3_lstm  <<<16,          256, 0, stream>>>(Gf, Gb, Wpad, Clen, H);
    k4a_att2 <<<128,         256, 0, stream>>>(H, Rh, att2);
    k4b_final<<<32,          256, 0, stream>>>(att2, H, Rh, simW, Clen, Rlen, dist, alldiff);
}